// MTL_Classifier_70755291234467
// MI455X (gfx1250) — compile-verified
//
#include <hip/hip_runtime.h>
#include <hip/hip_bf16.h>

// CRF NLL for B=256, T=8192, N_TAGS=16 on gfx1250.
//
// Forward recurrence kept in EXP SPACE with per-row power-of-2 exponent:
//   state: P[m,n] >= 0, c2[m];  true log2-alpha = c2[m] + log2(P[m,n])
//   step:  P <- (P @ E) * 2^(e_t * log2e)        (E = exp(trans), NEG -> 0)
//   renorm every 4 steps: ex = frexp_exp(P[m,0]); P *= 2^-ex; c2 += ex
// => per step: 4 chained V_WMMA_F32_16X16X4_F32 + cheap VALU, zero logs.
// Time loop manually blocked by 4 so renorm/prefetch are compile-time per
// sub-step (no per-iteration (t&3) branches, 4x amortized address math).

#define NTAG 16
#define SOS_T 14
#define EOS_T 15

typedef __attribute__((ext_vector_type(2))) float v2f;
typedef __attribute__((ext_vector_type(8))) float v8f;

#define LOG2E_F 1.4426950408889634f
#define LN2_F   0.6931471805599453f

// ds_swizzle on a float lane value (imm must be an ICE).
#define SWZ(x, imm) __int_as_float(__builtin_amdgcn_ds_swizzle(__float_as_int(x), (imm)))
// group-of-32 swizzle immediates: offset = xor[14:10] | or[9:5] | and[4:0]
#define SWZ_BCAST16 0x0010  // src = lane & 0x10 (broadcast lane 0/16 per half)
#define SWZ_XOR1    0x041F
#define SWZ_XOR2    0x081F
#define SWZ_XOR4    0x101F
#define SWZ_XOR8    0x201F

// ---------------------------------------------------------------------------
// Forward (partition) kernel.
// Block = 256 threads = 8 waves; each wave owns 16 consecutive batches.
// D/C layout: reg r on lane l (n=l&15, h=l>>4) holds
//   element [row r+8h][col n]  (row = batch-in-wave, col = tag).
// A-operand (f32 16x4) assumption: lane = M row, slice c pair .x/.y = K=4c+2h(+1).
// B-operand (f32 4x16) assumption mirrors A with lane = N col.
// ---------------------------------------------------------------------------
__global__ __launch_bounds__(256) void crf_forward_kernel(
    const float* __restrict__ em,    // [B,T,16]
    const float* __restrict__ mask,  // [B,T]
    const float* __restrict__ trans, // [16,16]
    float* __restrict__ partition,   // [B]
    int T)
{
  __shared__ float tile_all[8][16 * 17];  // per-wave transpose tile, padded
  const int lane = threadIdx.x & 31;
  const int wave = threadIdx.x >> 5;
  float* tile = tile_all[wave];
  const int n = lane & 15;  // tag column for this lane
  const int h = lane >> 4;  // lane half
  const int b0 = (blockIdx.x * 8 + wave) * 16;

  // Resident B operand: E = exp(trans) in 4 K-slices (NEG entries -> 0.0f).
  v2f vB[4];
#pragma unroll
  for (int c = 0; c < 4; ++c) {
    const int k0 = 4 * c + 2 * h;
    vB[c].x = __builtin_amdgcn_exp2f(trans[k0 * NTAG + n] * LOG2E_F);
    vB[c].y = __builtin_amdgcn_exp2f(trans[(k0 + 1) * NTAG + n] * LOG2E_F);
  }
  const float tSOS = trans[SOS_T * NTAG + n];  // trans[SOS][n]
  const float tEOS = trans[n * NTAG + EOS_T];  // trans[n][EOS]

  // Per-row global bases (row r of this lane-half -> batch b0+r+8h).
  size_t eb[8], mb[8];
#pragma unroll
  for (int r = 0; r < 8; ++r) {
    const size_t b = (size_t)(b0 + r + 8 * h);
    eb[r] = b * (size_t)T * NTAG + n;
    mb[r] = b * (size_t)T;
  }

  // Init: log2-alpha0 = (trans[SOS][n] + em[b,0,n]) * log2e.
  // Shift by tag-0 column (always finite) so P[m,0] = 1.
  float P[8], c2[8];
#pragma unroll
  for (int r = 0; r < 8; ++r) {
    const float v = (tSOS + em[eb[r]]) * LOG2E_F;
    const float c0 = SWZ(v, SWZ_BCAST16);
    c2[r] = c0;
    P[r] = __builtin_amdgcn_exp2f(v - c0);  // exp2(-huge) -> 0 for SOS/PAD cols
  }

  // One recurrence step; `renorm`/`pf` are compile-time constants at each
  // call site (inlined), so no per-iteration branches survive.
  auto step = [&](int t, bool renorm, bool pf) {
    // Emissions (as 2^(e*log2e)) + mask; off the critical recurrence path.
    float x[8], mk[8];
#pragma unroll
    for (int r = 0; r < 8; ++r) {
      x[r] = __builtin_amdgcn_exp2f(em[eb[r] + (size_t)t * NTAG] * LOG2E_F);
      mk[r] = mask[mb[r] + (size_t)t];
    }
    if (pf) {  // prefetch emissions ~32 steps ahead (global_prefetch_b8)
#pragma unroll
      for (int r = 0; r < 8; ++r)
        __builtin_prefetch(&em[eb[r] + (size_t)(t + 32) * NTAG], 0, 0);
    }

    // Stage P (D-layout) -> LDS -> reload as A operand (layout transpose).
#pragma unroll
    for (int r = 0; r < 8; ++r) tile[(r + 8 * h) * 17 + n] = P[r];
    asm volatile("s_wait_dscnt 0" ::: "memory");
    v2f vA[4];
#pragma unroll
    for (int c = 0; c < 4; ++c) {
      const int k0 = 4 * c + 2 * h;
      vA[c].x = tile[n * 17 + k0];
      vA[c].y = tile[n * 17 + k0 + 1];
    }

    // acc = P @ E : 4 chained f32 WMMAs over K=16.
    v8f acc = {};
#pragma unroll
    for (int c = 0; c < 4; ++c)
      acc = __builtin_amdgcn_wmma_f32_16x16x4_f32(
          false, vA[c], false, vB[c], (short)0, acc, false, false);

    // P_new = acc * 2^e ; masked select (v_cndmask, branch-free).
#pragma unroll
    for (int r = 0; r < 8; ++r) {
      const float upd = acc[r] * x[r];
      P[r] = (mk[r] != 0.0f) ? upd : P[r];
    }

    // Exponent-only renorm (no transcendentals, value-preserving, mask-safe).
    if (renorm) {
#pragma unroll
      for (int r = 0; r < 8; ++r) {
        const float bb = SWZ(P[r], SWZ_BCAST16);         // row's tag-0 value
        const int ex = __builtin_amdgcn_frexp_expf(bb);  // v_frexp_exp_i32_f32
        P[r] = __builtin_ldexpf(P[r], -ex);              // v_ldexp_f32
        c2[r] += (float)ex;
      }
    }
  };

  // Main loop blocked by 4: t = 1 + 4*i, renorm on the sub-step where
  // (t+k) % 4 == 0 (growth <= ~2^16/step, so 4-step window is safe in f32).
  int t = 1;
  for (; t + 3 < T - 32; t += 4) {
    step(t + 0, false, true);   // prefetch once per block of 4
    step(t + 1, false, false);
    step(t + 2, false, false);
    step(t + 3, true, false);   // (1+4i)+3 ≡ 0 (mod 4)
  }
  for (; t < T; ++t) step(t, (t & 3) == 0, false);

  // partition[b] = ln2 * (c2 + log2( sum_n P[m,n] * 2^(tEOS*log2e) )).
  const float w = __builtin_amdgcn_exp2f(tEOS * LOG2E_F);
#pragma unroll
  for (int r = 0; r < 8; ++r) {
    float s = P[r] * w;
    s += SWZ(s, SWZ_XOR1);
    s += SWZ(s, SWZ_XOR2);
    s += SWZ(s, SWZ_XOR4);
    s += SWZ(s, SWZ_XOR8);
    if (n == 0)
      partition[b0 + r + 8 * h] = (c2[r] + __builtin_amdgcn_logf(s)) * LN2_F;
  }
}

// ---------------------------------------------------------------------------
// Numerator (gold-path score) per batch: one block per batch, tree-reduced.
// ---------------------------------------------------------------------------
__global__ __launch_bounds__(256) void crf_score_kernel(
    const float* __restrict__ em, const int* __restrict__ tags,
    const float* __restrict__ mask, const float* __restrict__ trans,
    float* __restrict__ scores, int T)
{
  __shared__ float red[256];
  __shared__ float redm[256];
  const int b = blockIdx.x;
  const int tid = threadIdx.x;
  const size_t base = (size_t)b * T;
  float acc = 0.0f, msum = 0.0f;
  for (int t = tid; t < T; t += 256) {
    const int tg = tags[base + t];
    const float mk = mask[base + t];
    msum += mk;
    const float e = em[(base + t) * NTAG + tg];
    if (t == 0) {
      acc += trans[SOS_T * NTAG + tg] + e;  // unmasked first term
    } else {
      const int tp = tags[base + t - 1];
      acc += (e + trans[tp * NTAG + tg]) * mk;
    }
  }
  red[tid] = acc;
  redm[tid] = msum;
  __syncthreads();
  for (int s = 128; s > 0; s >>= 1) {
    if (tid < s) { red[tid] += red[tid + s]; redm[tid] += redm[tid + s]; }
    __syncthreads();
  }
  if (tid == 0) {
    const int last = (int)redm[0] - 1;
    const int lt = tags[base + last];
    scores[b] = red[0] + trans[lt * NTAG + EOS_T];
  }
}

// ---------------------------------------------------------------------------
// Final: out = sum_b (partition[b] - scores[b])   (== -sum(scores-partition))
// ---------------------------------------------------------------------------
__global__ __launch_bounds__(256) void crf_final_kernel(
    const float* __restrict__ partition, const float* __restrict__ scores,
    float* __restrict__ out, int B)
{
  __shared__ float red[256];
  const int tid = threadIdx.x;
  float a = 0.0f;
  for (int b = tid; b < B; b += 256) a += partition[b] - scores[b];
  red[tid] = a;
  __syncthreads();
  for (int s = 128; s > 0; s >>= 1) {
    if (tid < s) red[tid] += red[tid + s];
    __syncthreads();
  }
  if (tid == 0) out[0] = red[0];
}

extern "C" void kernel_launch(void* const* d_in, const int* in_sizes, int n_in,
                              void* d_out, int out_size, void* d_ws, size_t ws_size,
                              hipStream_t stream) {
  const float* em    = (const float*)d_in[0];  // [B,T,16] f32
  const int*   tags  = (const int*)d_in[1];    // [B,T] i32
  const float* mask  = (const float*)d_in[2];  // [B,T] f32
  const float* trans = (const float*)d_in[3];  // [16,16] f32

  const int B = 256;                 // per reference setup
  const int T = in_sizes[2] / B;     // mask is [B,T]

  float* scores = (float*)d_ws;      // [B]
  float* partition = scores + B;     // [B]

  crf_score_kernel<<<B, 256, 0, stream>>>(em, tags, mask, trans, scores, T);
  crf_forward_kernel<<<B / 128, 256, 0, stream>>>(em, mask, trans, partition, T);
  crf_final_kernel<<<1, 256, 0, stream>>>(partition, scores, (float*)d_out, B);
}